// LBPExtractor_39058432589917
// MI455X (gfx1250) — compile-verified
//
#include <hip/hip_runtime.h>
#include <stdint.h>

// ---------------- problem geometry ----------------
#define IMG_H   512
#define IMG_W   512
#define NBATCH  64
#define PLANE   (IMG_H * IMG_W)          // 262144 elems per channel plane
#define TILE    32                        // output tile edge
#define HALO    34                        // tile + 1-pixel edge halo
#define HELEMS  (HALO * HALO)             // 1156 logical halo elements
#define LROW    36                        // padded LDS row stride (16B-aligned rows)
#define LPAD_SLOT 1222u                   // pad-column slot for overflow lanes (33*36+34)
#define HPAD    1280                      // per-channel LDS staging extent (>= 34*36)
#define TILES_PER_IMG 256                 // (512/32)^2
#define TPW     4                         // tiles per workgroup (pipeline depth)
#define TPB     256                       // threads per block (8 wave32)
#define NWAVES  8
#define NBINS   256

// ---------------- CDNA5 async global->LDS path ----------------
#if defined(__gfx1250__) && __has_builtin(__builtin_amdgcn_global_load_async_to_lds_b32)
#define HAS_ASYNC_LDS 1
#else
#define HAS_ASYNC_LDS 0
#endif

typedef __attribute__((address_space(1))) int   gm_i32;   // global (device) int*
typedef __attribute__((address_space(3))) float lds_f32;  // LDS float*
typedef __attribute__((address_space(3))) int   lds_i32;  // LDS int*

__device__ __forceinline__ void copy_g2l_b32(const float* gsrc, float* ldst) {
#if HAS_ASYNC_LDS
    // async DMA: global -> LDS, tracked by ASYNCcnt, no VGPR landing.
    __builtin_amdgcn_global_load_async_to_lds_b32(
        (gm_i32*)(uintptr_t)gsrc, (lds_i32*)(lds_f32*)ldst, 0, 0);
#else
    *ldst = *gsrc;   // fallback: normal load + ds_store
#endif
}

template <int N>
__device__ __forceinline__ void wait_asynccnt() {
#if HAS_ASYNC_LDS
#if __has_builtin(__builtin_amdgcn_s_wait_asynccnt)
    __builtin_amdgcn_s_wait_asynccnt(N);
#else
    asm volatile("s_wait_asynccnt %0" :: "i"(N));
#endif
#endif
}

__global__ __launch_bounds__(TPB)
void lbp_hist_kernel(const float* __restrict__ x, uint32_t* __restrict__ ghist) {
    __shared__ float    stage[2][3][HPAD];         // double-buffered RGB staging
    __shared__ uint32_t hist[NWAVES][NBINS];       // per-wave private histograms

    const int tid   = threadIdx.x;
    const int wave  = tid >> 5;
    const int tile0 = blockIdx.x * TPW;            // 4 tiles, same image, same tile-row
    const int b     = tile0 >> 8;                  // 256 tiles per image
    const int lt0   = tile0 & (TILES_PER_IMG - 1);
    const int ty    = (lt0 >> 4) * TILE;           // constant tile-row for whole WG
    const int tx0   = (lt0 & 15) * TILE;

    const uint32_t base = (uint32_t)b * 3u * (uint32_t)PLANE;

    // ---- per-thread staging geometry, hoisted across all 4 tiles ----
    uint32_t lidx[5];     // LDS destination index within a channel plane
    uint32_t rowoff[5];   // clamp(ty-1+hy) * IMG_W  (row clamp is tile-invariant)
    int      hxs[5];      // halo column of this element
#pragma unroll
    for (int i = 0; i < 5; ++i) {
        const int e  = tid + i * TPB;              // 0..1279
        const int ec = (e < HELEMS) ? e : (HELEMS - 1);
        const int hy = ec / HALO;
        const int hx = ec - hy * HALO;
        int gy = ty - 1 + hy;
        gy = gy < 0 ? 0 : (gy > IMG_H - 1 ? IMG_H - 1 : gy);
        lidx[i]   = (e < HELEMS) ? (uint32_t)(hy * LROW + hx) : LPAD_SLOT;
        rowoff[i] = (uint32_t)(gy * IMG_W);
        hxs[i]    = hx;
    }

    // Issue one tile's RGB halo: exactly 5*3 = 15 async instructions per wave,
    // branch-free (overflow lanes write a clamped duplicate to the pad slot).
    auto stage_issue = [&](int tx, float* sbuf) {
#pragma unroll
        for (int i = 0; i < 5; ++i) {
            int gx = tx - 1 + hxs[i];
            gx = gx < 0 ? 0 : (gx > IMG_W - 1 ? IMG_W - 1 : gx);
            const uint32_t poff = base + rowoff[i] + (uint32_t)gx;
#pragma unroll
            for (int ch = 0; ch < 3; ++ch)
                copy_g2l_b32(x + poff + (uint32_t)ch * PLANE,
                             sbuf + ch * HPAD + lidx[i]);
        }
    };

    for (int i = tid; i < NWAVES * NBINS; i += TPB) (&hist[0][0])[i] = 0u;

    stage_issue(tx0, &stage[0][0][0]);             // prologue: tile 0
    __syncthreads();

    // Stencil geometry: thread handles 4 consecutive pixels of one row.
    const int srow = tid >> 3;                     // 0..31 output row
    const int scol = (tid & 7) * 4;                // 0,4,...,28 first output col

    for (int t = 0; t < TPW; ++t) {
        float* sbuf = &stage[t & 1][0][0];
        if (t + 1 < TPW) {                         // overlap next tile's staging
            stage_issue(tx0 + (t + 1) * TILE, &stage[(t & 1) ^ 1][0][0]);
            wait_asynccnt<15>();                   // leave next tile in flight
        } else {
            wait_asynccnt<0>();
        }
        __syncthreads();                           // all waves' copies landed

        // gray = .2989R + .587G + .114B, in place over the R plane
#pragma unroll
        for (int i = 0; i < 5; ++i) {
            if (tid + i * TPB < HELEMS) {
                const uint32_t li = lidx[i];
                const float r  = sbuf[li];
                const float g  = sbuf[HPAD + li];
                const float bl = sbuf[2 * HPAD + li];
                sbuf[li] = 0.2989f * r + 0.587f * g + 0.114f * bl;
            }
        }
        __syncthreads();                           // gray visible to all

        // 3 rows x 6 cols neighborhood via ds_load_b128 + ds_load_b64 per row
        const float* g0 = sbuf + srow * LROW + scol;           // 16B aligned
        const float4 a4 = *(const float4*)g0;
        const float2 a2 = *(const float2*)(g0 + 4);
        const float* g1 = g0 + LROW;
        const float4 b4 = *(const float4*)g1;
        const float2 b2 = *(const float2*)(g1 + 4);
        const float* g2 = g1 + LROW;
        const float4 c4 = *(const float4*)g2;
        const float2 c2 = *(const float2*)(g2 + 4);
        const float T[6] = {a4.x, a4.y, a4.z, a4.w, a2.x, a2.y};
        const float M[6] = {b4.x, b4.y, b4.z, b4.w, b2.x, b2.y};
        const float B[6] = {c4.x, c4.y, c4.z, c4.w, c2.x, c2.y};
        // OFFSETS: (-1,-1)b0 (-1,0)b1 (-1,1)b2 (0,1)b3 (1,1)b4 (1,0)b5 (1,-1)b6 (0,-1)b7
#pragma unroll
        for (int k = 0; k < 4; ++k) {
            const float c = M[k + 1];
            const uint32_t code =
                (T[k]     >= c ?   1u : 0u) | (T[k + 1] >= c ?   2u : 0u) |
                (T[k + 2] >= c ?   4u : 0u) | (M[k + 2] >= c ?   8u : 0u) |
                (B[k + 2] >= c ?  16u : 0u) | (B[k + 1] >= c ?  32u : 0u) |
                (B[k]     >= c ?  64u : 0u) | (M[k]     >= c ? 128u : 0u);
            atomicAdd(&hist[wave][code], 1u);      // ds_add_u32, wave-private
        }
        __syncthreads();                           // buf reusable next iter
    }

    uint32_t s = 0;
#pragma unroll
    for (int w = 0; w < NWAVES; ++w) s += hist[w][tid];
    if (s) atomicAdd(&ghist[b * NBINS + tid], s);  // one flush per WG per bin
}

// Per-batch L2 normalization: counts (u32, aliased in d_out) -> normalized f32.
__global__ __launch_bounds__(TPB)
void lbp_norm_kernel(float* __restrict__ io) {
    __shared__ float red[NBINS];
    const int b = blockIdx.x, t = threadIdx.x;
    const uint32_t cnt = ((const uint32_t*)io)[b * NBINS + t];
    const float v = (float)cnt;
    red[t] = v * v;
    __syncthreads();
#pragma unroll
    for (int s = NBINS / 2; s > 0; s >>= 1) {
        if (t < s) red[t] += red[t + s];
        __syncthreads();
    }
    io[b * NBINS + t] = v / (sqrtf(red[0]) + 1e-6f);
}

extern "C" void kernel_launch(void* const* d_in, const int* in_sizes, int n_in,
                              void* d_out, int out_size, void* d_ws, size_t ws_size,
                              hipStream_t stream) {
    (void)in_sizes; (void)n_in; (void)d_ws; (void)ws_size; (void)out_size;
    const float* x = (const float*)d_in[0];
    float* out = (float*)d_out;

    // d_out doubles as the u32 count buffer (64*256*4 B == out size). Zero it
    // every call (deterministic; capture-safe async memset).
    (void)hipMemsetAsync(out, 0, NBATCH * NBINS * sizeof(uint32_t), stream);

    const int nblocks = NBATCH * TILES_PER_IMG / TPW;    // 4096
    lbp_hist_kernel<<<nblocks, TPB, 0, stream>>>(x, (uint32_t*)out);
    lbp_norm_kernel<<<NBATCH, TPB, 0, stream>>>(out);
}